// GaussianStep_54597624266752
// MI455X (gfx1250) — compile-verified
//
#include <hip/hip_runtime.h>

// ---------------------------------------------------------------------------
// Batched Kalman step (update + predict) for G groups, S=16 state, M=8 meas.
// One wave32 per group; all 16x16x16 fp32 products on V_WMMA_F32_16X16X4_F32.
// Tile staging via GLOBAL_LOAD_ASYNC_TO_LDS_B128 (fallback: B128 vector copy).
// ---------------------------------------------------------------------------

typedef __attribute__((ext_vector_type(2))) float v2f;
typedef __attribute__((ext_vector_type(4))) float v4f;
typedef __attribute__((ext_vector_type(8))) float v8f;
typedef int v4i __attribute__((vector_size(16)));  // matches builtin pointee

#define HAS_ASYNC_LDS __has_builtin(__builtin_amdgcn_global_load_async_to_lds_b128)

namespace {

constexpr int S = 16;     // state dim
constexpr int M = 8;      // measurement dim
constexpr int WAVES = 4;  // waves (groups) per workgroup
constexpr int THREADS = WAVES * 32;

struct alignas(16) WaveScratch {
  float Hm[256];   // H zero-padded to 16x16 (rows 8..15 zero)
  float Cov[256];  // cov ; later reused to hold K@R
  float Rm[256];   // R zero-padded to 16x16
  float Fm[256];   // F
  float Km[256];   // K zero-padded to 16x16 (cols 8..15 zero)
  float T1[256];   // covHt ; later IKH ; later new_cov
  float T2[256];   // ikh@cov ; later F@new_cov
  float Aug[8 * 24];  // Gauss-Jordan augmented system [Ssys | U^T]
  float Mean[16];
  float Resid[8];
  float NewMean[16];
};

// Per-wave LDS ordering fence (waves never share LDS -> no block barrier).
__device__ __forceinline__ void lds_sync() {
  asm volatile("s_wait_dscnt 0" ::: "memory");
  __builtin_amdgcn_wave_barrier();
}

__device__ __forceinline__ void async_wait() {
#if HAS_ASYNC_LDS
#if __has_builtin(__builtin_amdgcn_s_wait_asynccnt)
  __builtin_amdgcn_s_wait_asynccnt(0);
#else
  asm volatile("s_wait_asynccnt 0" ::: "memory");
#endif
  __builtin_amdgcn_wave_barrier();
#endif
}

// acc += op(A) @ op(B) for 16x16 row-major fp32 tiles in LDS,
// chained over K in 4 steps of V_WMMA_F32_16X16X4_F32.
// A fragment: lane rc = row; VGPR0/1 = K = (k0 + 2*hi) + {0,1}.
// B fragment: lane rc = col; VGPR0/1 = K rows (k0 + 2*hi) + {0,1}.
template <bool TA, bool TB>
__device__ __forceinline__ v8f wmma_acc(const float* A, const float* B, v8f acc,
                                        int rc, int hi) {
#pragma unroll
  for (int k0 = 0; k0 < 16; k0 += 4) {
    const int ka = k0 + 2 * hi;
    v2f a, b;
    if constexpr (!TA) {
      a.x = A[rc * 16 + ka];
      a.y = A[rc * 16 + ka + 1];
    } else {
      a.x = A[ka * 16 + rc];
      a.y = A[(ka + 1) * 16 + rc];
    }
    if constexpr (!TB) {
      b.x = B[ka * 16 + rc];
      b.y = B[(ka + 1) * 16 + rc];
    } else {
      b.x = B[rc * 16 + ka];
      b.y = B[rc * 16 + ka + 1];
    }
    acc = __builtin_amdgcn_wmma_f32_16x16x4_f32(false, a, false, b, (short)0,
                                                acc, false, false);
  }
  return acc;
}

// C/D fragment: VGPR j holds row (j + 8*hi), column rc.
__device__ __forceinline__ void store_frag(float* T, v8f d, int rc, int hi) {
#pragma unroll
  for (int j = 0; j < 8; ++j) T[(j + 8 * hi) * 16 + rc] = d[j];
}

}  // namespace

__global__ __launch_bounds__(THREADS) void kalman_wmma_kernel(
    const float* __restrict__ g_in, const float* __restrict__ g_mean,
    const float* __restrict__ g_cov, const float* __restrict__ g_H,
    const float* __restrict__ g_R, const float* __restrict__ g_F,
    const float* __restrict__ g_Q, float* __restrict__ o_mean,
    float* __restrict__ o_cov, int G) {
  __shared__ WaveScratch ws[WAVES];
  const int wave = threadIdx.x >> 5;
  const int lane = threadIdx.x & 31;
  const int rc = lane & 15;  // row for A-frag, col for B/C/D-frags
  const int hi = lane >> 4;  // half-wave select
  const long g = (long)blockIdx.x * WAVES + wave;
  if (g >= G) return;  // wave-uniform; safe (no block barriers used)
  WaveScratch& w = ws[wave];

  const float* Hg = g_H + g * (M * S);
  const float* Rg = g_R + g * (M * M);
  const float* Cg = g_cov + g * (S * S);
  const float* Fg = g_F + g * (S * S);
  const float* Qg = g_Q + g * (S * S);

  // ---- zero pads (DS pipe; disjoint from async-loaded bytes) -----------
  {
    const v4f z4 = {};
    ((v4f*)w.Hm)[32 + lane] = z4;  // H rows 8..15
    ((v4f*)w.Rm)[lane] = z4;
    ((v4f*)w.Rm)[lane + 32] = z4;
    ((v4f*)w.Km)[lane] = z4;
    ((v4f*)w.Km)[lane + 32] = z4;
  }

  // ---- stage cov / F / H payload (1KB + 1KB + 512B per group) ----------
#if HAS_ASYNC_LDS
  {
    typedef __attribute__((address_space(1))) v4i gv4i;
    typedef __attribute__((address_space(3))) v4i lv4i;
    gv4i* cg = (gv4i*)Cg;
    gv4i* fg = (gv4i*)Fg;
    gv4i* hg = (gv4i*)Hg;
    lv4i* lc = (lv4i*)w.Cov;
    lv4i* lf = (lv4i*)w.Fm;
    lv4i* lh = (lv4i*)w.Hm;
    __builtin_amdgcn_global_load_async_to_lds_b128(cg + lane, lc + lane, 0, 0);
    __builtin_amdgcn_global_load_async_to_lds_b128(cg + lane + 32,
                                                   lc + lane + 32, 0, 0);
    __builtin_amdgcn_global_load_async_to_lds_b128(fg + lane, lf + lane, 0, 0);
    __builtin_amdgcn_global_load_async_to_lds_b128(fg + lane + 32,
                                                   lf + lane + 32, 0, 0);
    __builtin_amdgcn_global_load_async_to_lds_b128(hg + lane, lh + lane, 0, 0);
  }
#else
  {
    const v4f* CgV = (const v4f*)Cg;
    const v4f* FgV = (const v4f*)Fg;
    const v4f* HgV = (const v4f*)Hg;
    ((v4f*)w.Cov)[lane] = CgV[lane];
    ((v4f*)w.Cov)[lane + 32] = CgV[lane + 32];
    ((v4f*)w.Fm)[lane] = FgV[lane];
    ((v4f*)w.Fm)[lane + 32] = FgV[lane + 32];
    ((v4f*)w.Hm)[lane] = HgV[lane];  // rows 0..7 (same linearization)
  }
#endif

  // ---- R payload into padded rows (16B-aligned chunks) -----------------
  if (lane < 16) {
    const int row = lane >> 1, half = lane & 1;
    *(v4f*)(w.Rm + row * 16 + half * 4) =
        *(const v4f*)(Rg + row * M + half * 4);
  }
  if (lane < S) w.Mean[lane] = g_mean[g * S + lane];
  lds_sync();
  async_wait();

  // ---- U = cov @ H^T  (valid 16x8, cols 8..15 zero) --------------------
  v8f u = {};
  u = wmma_acc<false, true>(w.Cov, w.Hm, u, rc, hi);
  store_frag(w.T1, u, rc, hi);
  lds_sync();

  // ---- Ssys = H @ U + R  (valid 8x8) -----------------------------------
  v8f s;
#pragma unroll
  for (int j = 0; j < 8; ++j) s[j] = w.Rm[(j + 8 * hi) * 16 + rc];
  s = wmma_acc<false, false>(w.Hm, w.T1, s, rc, hi);

  // ---- Aug = [Ssys | U^T]  (8 x 24) ------------------------------------
  if (hi == 0 && rc < 8) {
#pragma unroll
    for (int j = 0; j < 8; ++j) w.Aug[j * 24 + rc] = s[j];
  }
#pragma unroll
  for (int i = lane; i < 128; i += 32) {
    const int p = i & 7, j = i >> 3;
    w.Aug[p * 24 + 8 + j] = w.T1[j * 16 + p];
  }
  lds_sync();

  // ---- Gauss-Jordan (SPD, no pivoting): Aug -> [I | K^T] ---------------
  for (int p = 0; p < 8; ++p) {
    const float pivinv = 1.0f / w.Aug[p * 24 + p];
    if (lane < 24) w.Aug[p * 24 + lane] *= pivinv;
    lds_sync();
#pragma unroll
    for (int r = 0; r < 8; ++r) {
      if (r == p) continue;
      const float f = w.Aug[r * 24 + p];
      if (lane < 24) w.Aug[r * 24 + lane] -= f * w.Aug[p * 24 + lane];
    }
    lds_sync();
  }

  // ---- K (16x8, zero-padded):  K[j][p] = Aug[p][8+j] -------------------
#pragma unroll
  for (int i = lane; i < 128; i += 32) {
    const int p = i & 7, j = i >> 3;
    w.Km[j * 16 + p] = w.Aug[p * 24 + 8 + j];
  }
  lds_sync();

  // ---- mean path (tiny; plain VALU) ------------------------------------
  if (lane < M) {
    float acc = g_in[g * M + lane];
#pragma unroll
    for (int k = 0; k < S; ++k) acc -= w.Hm[lane * 16 + k] * w.Mean[k];
    w.Resid[lane] = acc;
  }
  lds_sync();
  if (lane < S) {
    float nm = w.Mean[lane];
#pragma unroll
    for (int k = 0; k < M; ++k) nm += w.Km[lane * 16 + k] * w.Resid[k];
    w.NewMean[lane] = nm;
  }
  lds_sync();
  if (lane < S) {
    float pm = 0.0f;
#pragma unroll
    for (int k = 0; k < S; ++k) pm += w.Fm[lane * 16 + k] * w.NewMean[k];
    o_mean[g * S + lane] = pm;
  }

  // ---- ikh = I - K @ H --------------------------------------------------
  v8f kh = {};
  kh = wmma_acc<false, false>(w.Km, w.Hm, kh, rc, hi);
  v8f ikh;
#pragma unroll
  for (int j = 0; j < 8; ++j) {
    const int row = j + 8 * hi;
    ikh[j] = (row == rc ? 1.0f : 0.0f) - kh[j];
  }
  store_frag(w.T1, ikh, rc, hi);  // T1 := IKH
  lds_sync();

  // ---- T2 = ikh @ cov ---------------------------------------------------
  v8f t2 = {};
  t2 = wmma_acc<false, false>(w.T1, w.Cov, t2, rc, hi);
  store_frag(w.T2, t2, rc, hi);
  lds_sync();

  // ---- new_cov = T2 @ ikh^T + (K @ R) @ K^T  (Joseph form) --------------
  v8f nc = {};
  nc = wmma_acc<false, true>(w.T2, w.T1, nc, rc, hi);
  v8f kr = {};
  kr = wmma_acc<false, false>(w.Km, w.Rm, kr, rc, hi);
  store_frag(w.Cov, kr, rc, hi);  // Cov := K@R (cov dead)
  lds_sync();
  nc = wmma_acc<false, true>(w.Cov, w.Km, nc, rc, hi);
  store_frag(w.T1, nc, rc, hi);  // T1 := new_cov (IKH dead)
  lds_sync();

  // ---- pred_cov = F @ new_cov @ F^T + Q ---------------------------------
  v8f t3 = {};
  t3 = wmma_acc<false, false>(w.Fm, w.T1, t3, rc, hi);
  store_frag(w.T2, t3, rc, hi);
  lds_sync();
  v8f pc;
#pragma unroll
  for (int j = 0; j < 8; ++j) pc[j] = Qg[(j + 8 * hi) * 16 + rc];
  pc = wmma_acc<false, true>(w.T2, w.Fm, pc, rc, hi);

  float* oc = o_cov + g * (S * S);
#pragma unroll
  for (int j = 0; j < 8; ++j) oc[(j + 8 * hi) * 16 + rc] = pc[j];
}

extern "C" void kernel_launch(void* const* d_in, const int* in_sizes, int n_in,
                              void* d_out, int out_size, void* d_ws,
                              size_t ws_size, hipStream_t stream) {
  const float* g_in = (const float*)d_in[0];
  const float* g_mean = (const float*)d_in[1];
  const float* g_cov = (const float*)d_in[2];
  const float* g_H = (const float*)d_in[3];
  const float* g_R = (const float*)d_in[4];
  const float* g_F = (const float*)d_in[5];
  const float* g_Q = (const float*)d_in[6];
  const int G = in_sizes[0] / M;  // input is [G, M]
  float* o_mean = (float*)d_out;
  float* o_cov = o_mean + (size_t)G * S;

  dim3 grid((G + WAVES - 1) / WAVES), block(THREADS);
  hipLaunchKernelGGL(kalman_wmma_kernel, grid, block, 0, stream, g_in, g_mean,
                     g_cov, g_H, g_R, g_F, g_Q, o_mean, o_cov, G);
}